// NormalSmoothLoss_31928786878946
// MI455X (gfx1250) — compile-verified
//
#include <hip/hip_runtime.h>
#include <cstdint>

// ---------------------------------------------------------------------------
// NormalSmoothLoss for MI455X (gfx1250, wave32).
// Distance Gram matrix via V_WMMA_F32_16X16X4_F32 (K=4 >= 3 coords).
// Block = 8 waves = 2 query tiles x 4 column chunks (8192 waves total for
// machine fill). Per tile: 1 WMMA -> 16x16 d2 tile -> LDS transpose ->
// per-lane top-8 (branch-free cndmask bubble insert). Column point loads are
// software-pipelined one tile ahead to hide the global_load_b96 latency.
// ---------------------------------------------------------------------------

typedef float v2f __attribute__((ext_vector_type(2)));
typedef float v8f __attribute__((ext_vector_type(8)));

#define BATCH     4
#define NPTS      8192
#define KNBR      8
#define TILE      16
#define NTILES    (NPTS / TILE)   // 512
#define CHUNKS    4               // column-split factor (waves per query tile)
#define TPC       (NTILES / CHUNKS) // 128 tiles per chunk
#define QTPB      2               // query tiles per block
#define WAVES_PB  (QTPB * CHUNKS) // 8 waves, 256 threads
#define TSTRIDE   20              // padded LDS row stride: conflict-free, 16B-aligned rows

// 0.05 / (B * N * k * 3) -- fold WEIGHT and the mean divisor into the atomic add.
#define OUT_SCALE (0.05f / (float)(BATCH * NPTS * KNBR * 3))

// Sorted-ascending top-8 insert as a single-condition min/max bubble chain:
// per slot 1 v_cmp + 4 v_cndmask (no divergent branch tree). The early-out
// compare against the current 8th value keeps the common path at ~2 ops.
__device__ __forceinline__ void insert8(float v, int idx, float (&bv)[8], int (&bi)[8]) {
  if (v >= bv[7]) return;          // not in top-8 (also skips the +inf self marker)
#pragma unroll
  for (int s = 0; s < 8; ++s) {
    const bool  lt = v < bv[s];
    const float ov = bv[s];
    const int   oi = bi[s];
    bv[s] = lt ? v   : ov;         // keep the min in the slot
    bi[s] = lt ? idx : oi;
    v     = lt ? ov  : v;          // carry the max upward
    idx   = lt ? oi  : idx;
  }
}

__device__ __forceinline__ float3 load3(const float* __restrict__ p) {
  return make_float3(p[0], p[1], p[2]);   // merges into one global_load_b96
}

__global__ __launch_bounds__(WAVES_PB * 32)
void NormalSmoothLoss_kernel(const float* __restrict__ points,
                             const float* __restrict__ normals,
                             float* __restrict__ out) {
  // Per-wave 16x16 transpose scratch, padded rows (stride 20 words):
  // low-half lanes hit banks (20r+n)%64, high-half lanes are offset by 32 banks.
  __shared__ float tbuf[WAVES_PB][TILE * TSTRIDE];
  // Cross-wave merge buffers: per query tile, per row, per chunk, top-8.
  __shared__ float mv[QTPB][TILE][CHUNKS][KNBR];
  __shared__ int   mi[QTPB][TILE][CHUNKS][KNBR];

  const int  lane  = threadIdx.x & 31;
  const int  wave  = threadIdx.x >> 5;
  const bool hi    = lane >= 16;          // upper half-wave
  const int  l16   = lane & 15;
  const int  qtl   = wave >> 2;           // query tile within block (0..1)
  const int  chunk = wave & 3;            // column chunk (0..3)

  const int qt    = blockIdx.x * QTPB + qtl;      // global query-tile id
  const int b     = qt / NTILES;
  const int qbase = (qt % NTILES) * TILE;

  const float* pts = points  + (size_t)b * NPTS * 3;
  const float* nrm = normals + (size_t)b * NPTS * 3;

  // ---- Query tile -> WMMA A operand (pre-scaled by -2) --------------------
  // A 16x4 layout (ISA 7.12.2): lanes 0-15 hold K0,K1; lanes 16-31 hold K2,K3.
  const int    myq = qbase + l16;
  const float3 qp  = load3(pts + myq * 3);
  const float  sqq = qp.x * qp.x + qp.y * qp.y + qp.z * qp.z;

  v2f a;
  a.x = -2.0f * (hi ? qp.z : qp.x);   // K2 : K0
  a.y = -2.0f * (hi ? 0.0f : qp.y);   // K3 : K1

  // ---- C operand: |p_q|^2 broadcast into the C/D row layout ---------------
  // C/D layout: VGPR i -> row i (lanes 0-15) / row i+8 (lanes 16-31).
  const int rowoff = hi ? 8 : 0;
  v8f c;
#pragma unroll
  for (int i = 0; i < 8; ++i) c[i] = __shfl(sqq, i + rowoff, 32);

  // ---- per-lane top-8 state: lane r -> row r, cols c%16<8; lane r+16 -> c%16>=8
  float bv[KNBR]; int bi[KNBR];
#pragma unroll
  for (int j = 0; j < KNBR; ++j) { bv[j] = __builtin_inff(); bi[j] = 0; }

  const int   myrow = l16;
  const int   coff  = hi ? 8 : 0;
  float* const tb   = &tbuf[wave][0];

  // ---- scan this wave's 128 column tiles, point load pipelined 1 ahead ----
  const int tc0 = chunk * TPC;
  float3 cp = load3(pts + (tc0 * TILE + l16) * 3);
  for (int t = tc0; t < tc0 + TPC; ++t) {
    const int cbase = t * TILE;
    const int tn    = (t + 1 < tc0 + TPC) ? (t + 1) : t;   // clamped prefetch
    const float3 np = load3(pts + (tn * TILE + l16) * 3);

    const float sqc = cp.x * cp.x + cp.y * cp.y + cp.z * cp.z;

    // B 4x16 operand, assumed analogous to C/D split: VGPR i -> K=i / K=i+2.
    v2f bm;
    bm.x = hi ? cp.z : cp.x;        // K2 : K0
    bm.y = hi ? 0.0f : cp.y;        // K3 : K1

    // D[m][n] = |p_m|^2 - 2 p_m.p_n   (one v_wmma_f32_16x16x4_f32)
    v8f d = __builtin_amdgcn_wmma_f32_16x16x4_f32(
        false, a, false, bm, (short)0, c, false, false);

    // d2 = D + |p_n|^2, stored transposed into LDS (row-major 16x16, stride 20).
#pragma unroll
    for (int i = 0; i < 8; ++i)
      tb[(i + rowoff) * TSTRIDE + l16] = d[i] + sqc;

    // DS ops of one wave execute in order (ISA 7.3): no explicit wait needed
    // for store->load visibility; just pin the compiler's program order.
    asm volatile("" ::: "memory");

    // Each lane reads its half-row (16B-aligned ds_load_b128 x2); the
    // compiler inserts the s_wait_dscnt before first use of the data.
    const float4* rp = (const float4*)(tb + myrow * TSTRIDE + coff);
    const float4 r0 = rp[0];
    const float4 r1 = rp[1];
    const float cand[8] = {r0.x, r0.y, r0.z, r0.w, r1.x, r1.y, r1.z, r1.w};

#pragma unroll
    for (int j = 0; j < 8; ++j) {
      const int colj = cbase + coff + j;
      const float v = (colj == qbase + myrow) ? __builtin_inff() : cand[j];
      insert8(v, colj, bv, bi);
    }
    asm volatile("" ::: "memory");  // keep next tile's stores after these loads
    cp = np;
  }

  // ---- merge partner half-row (lane ^ 16) ---------------------------------
#pragma unroll
  for (int j = 0; j < KNBR; ++j) {
    const float pv = __shfl_xor(bv[j], 16, 32);
    const int   pi = __shfl_xor(bi[j], 16, 32);
    insert8(pv, pi, bv, bi);        // lanes 0-15 now hold this chunk's row top-8
  }

  // ---- publish per-chunk results, merge chunks ----------------------------
  if (!hi) {
#pragma unroll
    for (int j = 0; j < KNBR; ++j) {
      mv[qtl][l16][chunk][j] = bv[j];
      mi[qtl][l16][chunk][j] = bi[j];
    }
  }
  __syncthreads();

  float s = 0.0f;
  if (chunk == 0 && !hi) {
    // bv/bi already hold chunk 0's top-8; fold in chunks 1..3.
#pragma unroll
    for (int cc = 1; cc < CHUNKS; ++cc)
#pragma unroll
      for (int j = 0; j < KNBR; ++j)
        insert8(mv[qtl][l16][cc][j], mi[qtl][l16][cc][j], bv, bi);

    // ---- gather neighbor normals, accumulate squared diff -----------------
    const float3 n0 = load3(nrm + myq * 3);
#pragma unroll
    for (int j = 0; j < KNBR; ++j) {
      const int nb = bi[j];
      const float dx = n0.x - nrm[nb * 3 + 0];
      const float dy = n0.y - nrm[nb * 3 + 1];
      const float dz = n0.z - nrm[nb * 3 + 2];
      s += dx * dx + dy * dy + dz * dz;
    }
  }

  // ---- wave butterfly reduce, one atomic per contributing wave ------------
#pragma unroll
  for (int off = 16; off > 0; off >>= 1) s += __shfl_xor(s, off, 32);
  if (chunk == 0 && lane == 0) atomicAdd(out, s * OUT_SCALE);
}

extern "C" void kernel_launch(void* const* d_in, const int* in_sizes, int n_in,
                              void* d_out, int out_size, void* d_ws, size_t ws_size,
                              hipStream_t stream) {
  const float* points  = (const float*)d_in[0];
  const float* normals = (const float*)d_in[1];
  // d_in[2] = k_neighbors (int scalar, fixed at 8 by the reference setup)
  (void)in_sizes; (void)n_in; (void)d_ws; (void)ws_size; (void)out_size;

  // Harness poisons d_out; zero it (capture-safe memset node) before atomics.
  hipMemsetAsync(d_out, 0, sizeof(float), stream);

  const int blocks = (BATCH * NTILES) / QTPB;   // 1024
  NormalSmoothLoss_kernel<<<blocks, WAVES_PB * 32, 0, stream>>>(
      points, normals, (float*)d_out);
}